// ConditionalLogisticRegression_56624848830665
// MI455X (gfx1250) — compile-verified
//
#include <hip/hip_runtime.h>
#include <hip/hip_bf16.h>

// ---------------------------------------------------------------------------
// ConditionalLogisticRegression on gfx1250 (MI455X)
//   logits = X @ W + b        -> fp32 WMMA (V_WMMA_F32_16X16X4_F32)
//   per-segment softmax       -> encoded-uint atomicMax + atomicAdd passes
// Memory-bound (~1.1 GiB traffic, ~50us at 23.3 TB/s); WMMA is free under
// the HBM roof and keeps full fp32 precision of the reference.
// ---------------------------------------------------------------------------

typedef float v2f __attribute__((ext_vector_type(2)));
typedef float v8f __attribute__((ext_vector_type(8)));

#define CLR_N        4194304
#define CLR_D        64
#define CLR_NSEG     131072
#define TILES_PER_WAVE 4           // 16 rows per tile, 4 tiles per wave

// Monotone order-preserving float<->uint key (works for negative values).
__device__ __forceinline__ unsigned f2key(float f) {
    unsigned u = __float_as_uint(f);
    return (u & 0x80000000u) ? ~u : (u | 0x80000000u);
}
__device__ __forceinline__ float key2f(unsigned k) {
    return (k & 0x80000000u) ? __uint_as_float(k ^ 0x80000000u)
                             : __uint_as_float(~k);
}

// Pass 0: initialize per-segment scratch (key 0 == -inf sentinel, sum 0).
__global__ void clr_init_seg_kernel(unsigned* __restrict__ segMaxKey,
                                    float* __restrict__ segSum, int nseg) {
    int i = blockIdx.x * blockDim.x + threadIdx.x;
    if (i < nseg) { segMaxKey[i] = 0u; segSum[i] = 0.0f; }
}

// Pass 1: logits via fp32 WMMA + per-segment max (encoded atomicMax).
// One wave (32 lanes) computes TILES_PER_WAVE tiles of 16 rows each.
__global__ void clr_logits_wmma_kernel(const float* __restrict__ X,
                                       const int*   __restrict__ seg,
                                       const float* __restrict__ W,
                                       const float* __restrict__ b,
                                       float*       __restrict__ logits,
                                       unsigned*    __restrict__ segMaxKey) {
    const int lane  = threadIdx.x & 31;
    const int half  = lane >> 4;          // 0: lanes 0-15, 1: lanes 16-31
    const int mrow  = lane & 15;          // A-matrix row index within tile
    const int waveInBlock  = threadIdx.x >> 5;
    const int wavesPerBlk  = blockDim.x >> 5;
    const long waveId = (long)blockIdx.x * wavesPerBlk + waveInBlock;

    const float bias = b[0];

    // Preload W fragments: B[k][n] = W[k] replicated across all 16 columns.
    // Lane L, VGPR v of B holds row (kbase + v + 2*half) -> W[kbase + v + 2*half].
    v2f wf[16];
#pragma unroll
    for (int ks = 0; ks < 16; ++ks) {
        const float* wp = W + 4 * ks + 2 * half;
        wf[ks].x = wp[0];
        wf[ks].y = wp[1];
    }

    for (int t = 0; t < TILES_PER_WAVE; ++t) {
        const long tile    = waveId * TILES_PER_WAVE + t;
        const long rowBase = tile * 16;
        // A layout: lane L, VGPR v -> X[rowBase + (L&15)][kbase + v + 2*half]
        const float* xrow = X + (rowBase + mrow) * CLR_D + 2 * half;

        v8f c = {0.f, 0.f, 0.f, 0.f, 0.f, 0.f, 0.f, 0.f};
#pragma unroll
        for (int ks = 0; ks < 16; ++ks) {
            v2f a = *(const v2f*)(xrow + 4 * ks);   // 8B-aligned float2
            // D = A(16x4) * B(4x16) + C ; fp32, RNE, EXEC all-ones here.
            c = __builtin_amdgcn_wmma_f32_16x16x4_f32(
                    /*neg_a=*/false, a, /*neg_b=*/false, wf[ks],
                    /*c_mod=*/(short)0, c, /*reuse_a=*/false, /*reuse_b=*/false);
        }

        // Every column of D equals the dot product; lane L holds rows
        // r + 8*half in c[r]. Lanes 0 and 16 emit all 16 row results.
        if ((lane & 15) == 0) {
#pragma unroll
            for (int r = 0; r < 8; ++r) {
                const long row = rowBase + 8 * half + r;
                const float lg = c[r] + bias;
                logits[row] = lg;
                atomicMax(&segMaxKey[seg[row]], f2key(lg));
            }
        }
    }
}

// Pass 2: e = exp(logit - segmax); write e to out; accumulate segment sums.
__global__ void clr_exp_sum_kernel(const float*    __restrict__ logits,
                                   const int*      __restrict__ seg,
                                   const unsigned* __restrict__ segMaxKey,
                                   float*          __restrict__ segSum,
                                   float*          __restrict__ outE, int n) {
    int i = blockIdx.x * blockDim.x + threadIdx.x;
    if (i < n) {
        const int   s = seg[i];
        const float m = key2f(segMaxKey[s]);
        const float e = __expf(logits[i] - m);
        outE[i] = e;
        atomicAdd(&segSum[s], e);
    }
}

// Pass 3: normalize in place.
__global__ void clr_norm_kernel(const int*   __restrict__ seg,
                                const float* __restrict__ segSum,
                                float*       __restrict__ out, int n) {
    int i = blockIdx.x * blockDim.x + threadIdx.x;
    if (i < n) out[i] = out[i] / segSum[seg[i]];
}

extern "C" void kernel_launch(void* const* d_in, const int* in_sizes, int n_in,
                              void* d_out, int out_size, void* d_ws, size_t ws_size,
                              hipStream_t stream) {
    const float* X   = (const float*)d_in[0];   // [N, 64] fp32
    const int*   seg = (const int*)  d_in[1];   // [N] int32, sorted
    const float* W   = (const float*)d_in[2];   // [64] fp32
    const float* b   = (const float*)d_in[3];   // [1] fp32
    float* out = (float*)d_out;                 // [N] fp32

    // Workspace carve-up: logits (N*4) | segMaxKey (NSEG*4) | segSum (NSEG*4)
    char* ws = (char*)d_ws;
    float*    logits    = (float*)ws;
    unsigned* segMaxKey = (unsigned*)(ws + (size_t)CLR_N * 4);
    float*    segSum    = (float*)(ws + (size_t)CLR_N * 4 + (size_t)CLR_NSEG * 4);

    // Pass 0: init segment scratch (must run every call; ws not re-poisoned).
    clr_init_seg_kernel<<<CLR_NSEG / 256, 256, 0, stream>>>(segMaxKey, segSum, CLR_NSEG);

    // Pass 1: WMMA logits + segment max.
    // tiles = N/16 = 262144; waves = tiles/TILES_PER_WAVE = 65536;
    // 8 waves per 256-thread block -> 8192 blocks.
    const int tiles  = CLR_N / 16;
    const int waves  = tiles / TILES_PER_WAVE;
    const int blocks = waves / 8;
    clr_logits_wmma_kernel<<<blocks, 256, 0, stream>>>(X, seg, W, b, logits, segMaxKey);

    // Pass 2: exp + segment sum.
    clr_exp_sum_kernel<<<CLR_N / 256, 256, 0, stream>>>(logits, seg, segMaxKey,
                                                        segSum, out, CLR_N);

    // Pass 3: normalize.
    clr_norm_kernel<<<CLR_N / 256, 256, 0, stream>>>(seg, segSum, out, CLR_N);
}